// ByteMulFFN_7945689497940
// MI455X (gfx1250) — compile-verified
//
#include <hip/hip_runtime.h>
#include <stdint.h>

// ---------------------------------------------------------------------------
// ByteMulFFN for MI455X (gfx1250).
//
// Memory-bound: 128 MiB in + 128 MiB out @ 23.3 TB/s => ~11.5 us floor; no
// matrix math => WMMA not applicable. Strategy: CDNA5 async Global<->LDS DMA
// (ASYNCcnt) turns the strided one-thread-per-row pattern into coalesced
// 512B-per-wave-instruction streams; argmax/gather/+2.0 happen against LDS.
// LDS rows padded to 528B (4-bank lane stagger) => only the unavoidable
// 2-way conflict on 16B/lane reads.
// ---------------------------------------------------------------------------

#define THREADS          64          // 2 waves (wave32)
#define ROWS_PER_BLOCK   64
#define ROW_FLOATS       128
#define ROW_BYTES        512
#define CHUNKS_PER_ROW   32          // 32 x b128 per row
#define TILE_CHUNKS      (ROWS_PER_BLOCK * CHUNKS_PER_ROW)   // 2048
#define LDS_ROW_STRIDE   528
#define LDS_BYTES        (ROWS_PER_BLOCK * LDS_ROW_STRIDE)   // 33792 B

// Byte-field layout constants (match the reference)
#define F_MARK_AX      0
#define F_OP_MUL       1
#define F_ALU_LO       2
#define F_ALU_HI       18
#define F_AXC_LO       34
#define F_AXC_HI       50
#define F_OUT_LO       66
#define F_OUT_HI       82

// ---- CDNA5 async Global<->LDS ops. LDS operand is a byte offset from the
// wave's LDS base; with one static __shared__ object our offsets start at 0.
__device__ __forceinline__ void async_load_b128(uint32_t lds_off, const void* gaddr) {
    asm volatile("global_load_async_to_lds_b128 %0, %1, off"
                 :: "v"(lds_off), "v"(gaddr) : "memory");
}
__device__ __forceinline__ void async_store_b128(void* gaddr, uint32_t lds_off) {
    asm volatile("global_store_async_from_lds_b128 %0, %1, off"
                 :: "v"(gaddr), "v"(lds_off) : "memory");
}
__device__ __forceinline__ void wait_async_zero() {
    asm volatile("s_wait_asynccnt 0" ::: "memory");
}

// First-occurrence argmax over 16 floats (matches jnp.argmax tie-break).
__device__ __forceinline__ int argmax16(const float* __restrict__ p) {
    float best = p[0];
    int   bi   = 0;
#pragma unroll
    for (int i = 1; i < 16; ++i) {
        float v = p[i];
        if (v > best) { best = v; bi = i; }
    }
    return bi;
}

__global__ __launch_bounds__(THREADS)
void ByteMulFFN_kernel(const float* __restrict__ x,
                       const float* __restrict__ tab,   // 256x256 f32, L2-hot
                       float* __restrict__ out,
                       int total_rows)
{
    __shared__ unsigned char smem[LDS_BYTES];

    const int       tid      = threadIdx.x;
    const long long row0     = (long long)blockIdx.x * ROWS_PER_BLOCK;
    const int       rowsHere = (total_rows - row0) < ROWS_PER_BLOCK
                                   ? (int)(total_rows - row0) : ROWS_PER_BLOCK;
    const int       chunks   = rowsHere * CHUNKS_PER_ROW;
    const bool      fullTile = (chunks == TILE_CHUNKS);

    const char* gin  = (const char*)(x + row0 * ROW_FLOATS);
    char*       gout = (char*)(out + row0 * ROW_FLOATS);

    // Starting coordinates for the copy loops: lane covers chunk (row, ch)
    // with row = tid>>5, ch = tid&31; each k-step advances 64 chunks = 2 rows.
    const int      r0      = tid >> 5;
    const int      c0      = (tid & 31) * 16;
    const uint32_t laBase  = (uint32_t)(r0 * LDS_ROW_STRIDE + c0);
    const int      gBase   = r0 * ROW_BYTES + c0;

    // ---- Phase A: coalesced async copy Global -> LDS ----------------------
    if (fullTile) {
        uint32_t    la = laBase;
        const char* ga = gin + gBase;
#pragma unroll
        for (int k = 0; k < CHUNKS_PER_ROW; ++k) {
            async_load_b128(la, (const void*)ga);
            la += 2 * LDS_ROW_STRIDE;
            ga += 2 * ROW_BYTES;
        }
    } else {                               // cold remainder path
        for (int k = 0; k < CHUNKS_PER_ROW; ++k) {
            int c = tid + k * THREADS;
            if (c < chunks) {
                uint32_t la = (uint32_t)((c >> 5) * LDS_ROW_STRIDE + (c & 31) * 16);
                async_load_b128(la, (const void*)(gin + (size_t)c * 16));
            }
        }
    }
    wait_async_zero();     // drain this wave's ASYNCcnt
    __syncthreads();       // make all waves' LDS tiles visible

    // ---- Phase B: per-row decode + table gather + in-LDS scatter-add ------
    if (tid < rowsHere) {
        float* rowf = (float*)(smem + tid * LDS_ROW_STRIDE);

        const bool mask = (rowf[F_MARK_AX] >= 0.5f) && (rowf[F_OP_MUL] >= 0.5f);

        const int a_lo = argmax16(rowf + F_ALU_LO);
        const int a_hi = argmax16(rowf + F_ALU_HI);
        const int b_lo = argmax16(rowf + F_AXC_LO);
        const int b_hi = argmax16(rowf + F_AXC_HI);
        const int a    = a_lo | (a_hi << 4);
        const int b    = b_lo | (b_hi << 4);

        // 256KB table stays resident in the 192MB L2; one gather per row.
        const int res  = (int)tab[a * 256 + b];

        if (mask) {
            // Disjoint windows [66,81] / [82,97]; row owned by this thread.
            rowf[F_OUT_LO + (res & 15)]        += 2.0f;
            rowf[F_OUT_HI + ((res >> 4) & 15)] += 2.0f;
        }
    }
    __syncthreads();       // LDS updates visible before the async drain

    // ---- Phase C: coalesced async copy LDS -> Global ----------------------
    if (fullTile) {
        uint32_t la = laBase;
        char*    ga = gout + gBase;
#pragma unroll
        for (int k = 0; k < CHUNKS_PER_ROW; ++k) {
            async_store_b128((void*)ga, la);
            la += 2 * LDS_ROW_STRIDE;
            ga += 2 * ROW_BYTES;
        }
    } else {
        for (int k = 0; k < CHUNKS_PER_ROW; ++k) {
            int c = tid + k * THREADS;
            if (c < chunks) {
                uint32_t la = (uint32_t)((c >> 5) * LDS_ROW_STRIDE + (c & 31) * 16);
                async_store_b128((void*)(gout + (size_t)c * 16), la);
            }
        }
    }
    wait_async_zero();     // (s_endpgm also drains, but be explicit)
}

extern "C" void kernel_launch(void* const* d_in, const int* in_sizes, int n_in,
                              void* d_out, int out_size, void* d_ws, size_t ws_size,
                              hipStream_t stream) {
    const float* x   = (const float*)d_in[0];   // [B,T,S] f32, S=128
    const float* tab = (const float*)d_in[1];   // [256,256] f32
    float*       out = (float*)d_out;

    const int total_rows = in_sizes[0] / ROW_FLOATS;        // B*T = 262144
    const int grid       = (total_rows + ROWS_PER_BLOCK - 1) / ROWS_PER_BLOCK;

    ByteMulFFN_kernel<<<grid, THREADS, 0, stream>>>(x, tab, out, total_rows);
}